// ThreeBodyInteractions_59442347376884
// MI455X (gfx1250) — compile-verified
//
#include <hip/hip_runtime.h>

typedef __attribute__((ext_vector_type(2))) float v2f;
typedef __attribute__((ext_vector_type(8))) float v8f;

#define FEAT 64
#define NBASIS 9

// ---------------------------------------------------------------------------
// Kernel 1: atoms = sigmoid(node_feat @ W_atom + b_atom)   [n_nodes, 9]
// ---------------------------------------------------------------------------
__global__ void k_atom_gate(const float* __restrict__ node_feat,
                            const float* __restrict__ W_atom,   // [64,9] row-major
                            const float* __restrict__ b_atom,   // [9]
                            float* __restrict__ atoms,          // [n_nodes,9]
                            int n_nodes) {
  __shared__ float sW[FEAT * NBASIS];
  __shared__ float sB[NBASIS];
  for (int i = threadIdx.x; i < FEAT * NBASIS; i += blockDim.x) sW[i] = W_atom[i];
  if (threadIdx.x < NBASIS) sB[threadIdx.x] = b_atom[threadIdx.x];
  __syncthreads();

  int n = blockIdx.x * blockDim.x + threadIdx.x;
  if (n >= n_nodes) return;

  float acc[NBASIS];
#pragma unroll
  for (int c = 0; c < NBASIS; ++c) acc[c] = sB[c];

  const float* row = node_feat + (size_t)n * FEAT;
#pragma unroll 4
  for (int f = 0; f < FEAT; ++f) {
    float x = row[f];
#pragma unroll
    for (int c = 0; c < NBASIS; ++c) acc[c] = fmaf(x, sW[f * NBASIS + c], acc[c]);
  }
  float* o = atoms + (size_t)n * NBASIS;
#pragma unroll
  for (int c = 0; c < NBASIS; ++c) o[c] = 1.0f / (1.0f + __expf(-acc[c]));
}

// ---------------------------------------------------------------------------
// Kernel 2: per-edge segmented reduction over sorted segment_ids.
//   nb[s][c] = sum_{t: segment_ids[t]==s} three_basis[t][c] * atoms[edge_dst[lg_dst[t]]][c]
// No atomics: one thread per segment, binary-search its run.
// ---------------------------------------------------------------------------
__device__ __forceinline__ int lower_bound_i32(const int* __restrict__ a, int n, int key) {
  int lo = 0, hi = n;
  while (lo < hi) {
    int mid = (lo + hi) >> 1;
    if (a[mid] < key) lo = mid + 1; else hi = mid;
  }
  return lo;
}

__global__ void k_segsum(const float* __restrict__ three_basis,  // [n_triples,9]
                         const float* __restrict__ atoms,        // [n_nodes,9]
                         const int* __restrict__ edge_dst,       // [n_edges]
                         const int* __restrict__ lg_dst,         // [n_triples]
                         const int* __restrict__ segment_ids,    // [n_triples] sorted
                         float* __restrict__ nb,                 // [n_edges,9]
                         int n_edges, int n_triples) {
  int s = blockIdx.x * blockDim.x + threadIdx.x;
  if (s >= n_edges) return;

  int lo = lower_bound_i32(segment_ids, n_triples, s);
  int hi = lower_bound_i32(segment_ids, n_triples, s + 1);

  float acc[NBASIS];
#pragma unroll
  for (int c = 0; c < NBASIS; ++c) acc[c] = 0.0f;

  for (int t = lo; t < hi; ++t) {
    int e = lg_dst[t];
    int a = edge_dst[e];
    const float* ar = atoms + (size_t)a * NBASIS;
    const float* br = three_basis + (size_t)t * NBASIS;
#pragma unroll
    for (int c = 0; c < NBASIS; ++c) acc[c] = fmaf(br[c], ar[c], acc[c]);
  }
  float* o = nb + (size_t)s * NBASIS;
#pragma unroll
  for (int c = 0; c < NBASIS; ++c) o[c] = acc[c];
}

// ---------------------------------------------------------------------------
// Kernel 3: out = edge_feat + silu(NB@W_out + b_out) * sigmoid(NB@W_gate + b_gate)
// One wave32 per 16-edge tile; V_WMMA_F32_16X16X4_F32, K=9 padded to 12
// (3 K-steps), N=64 as 4 tiles of 16. A fragments shared by both GEMMs.
// ---------------------------------------------------------------------------
__global__ void __launch_bounds__(256)
k_bond_update(const float* __restrict__ nb,        // [n_edges,9]
              const float* __restrict__ edge_feat, // [n_edges,64]
              const float* __restrict__ W_out,     // [9,64]
              const float* __restrict__ b_out,     // [64]
              const float* __restrict__ W_gate,    // [9,64]
              const float* __restrict__ b_gate,    // [64]
              float* __restrict__ out,             // [n_edges,64]
              int n_edges) {
  const int lane = threadIdx.x & 31;
  const int wave = threadIdx.x >> 5;
  const int tile = blockIdx.x * (blockDim.x >> 5) + wave;
  const int edge0 = tile * 16;
  if (edge0 >= n_edges) return;  // wave-uniform: EXEC stays all-ones for WMMA

  const int half = lane >> 4;   // selects K sub-pair per ISA 32-bit A/B layout
  const int nl   = lane & 15;   // M for A, N for B/C/D

  // ---- A fragments (16x4 f32 per step): lanes 0-15 K={4k,4k+1}, 16-31 K={4k+2,4k+3}
  v2f A[3];
#pragma unroll
  for (int k = 0; k < 3; ++k) {
    int k0 = 4 * k + 2 * half;
    const float* arow = nb + (size_t)(edge0 + nl) * NBASIS;
    A[k].x = (k0 + 0 < NBASIS) ? arow[k0 + 0] : 0.0f;
    A[k].y = (k0 + 1 < NBASIS) ? arow[k0 + 1] : 0.0f;
  }

  // ---- B fragments (4x16 f32 per step) for both weight matrices, all 4 N-tiles
  v2f Bo[4][3], Bg[4][3];
#pragma unroll
  for (int nt = 0; nt < 4; ++nt) {
    int n = nl + nt * 16;
#pragma unroll
    for (int k = 0; k < 3; ++k) {
      int k0 = 4 * k + 2 * half;
      Bo[nt][k].x = (k0 + 0 < NBASIS) ? W_out [(k0 + 0) * FEAT + n] : 0.0f;
      Bo[nt][k].y = (k0 + 1 < NBASIS) ? W_out [(k0 + 1) * FEAT + n] : 0.0f;
      Bg[nt][k].x = (k0 + 0 < NBASIS) ? W_gate[(k0 + 0) * FEAT + n] : 0.0f;
      Bg[nt][k].y = (k0 + 1 < NBASIS) ? W_gate[(k0 + 1) * FEAT + n] : 0.0f;
    }
  }

#pragma unroll
  for (int nt = 0; nt < 4; ++nt) {
    v8f co = {};
    v8f cg = {};
#pragma unroll
    for (int k = 0; k < 3; ++k) {
      co = __builtin_amdgcn_wmma_f32_16x16x4_f32(false, A[k], false, Bo[nt][k],
                                                 (short)0, co, false, false);
      cg = __builtin_amdgcn_wmma_f32_16x16x4_f32(false, A[k], false, Bg[nt][k],
                                                 (short)0, cg, false, false);
    }
    const int n = nl + nt * 16;
    const float bo = b_out[n];
    const float bg = b_gate[n];
#pragma unroll
    for (int v = 0; v < 8; ++v) {
      int m = v + half * 8;                       // C/D layout: row = VGPR + 8*(lane>=16)
      size_t idx = (size_t)(edge0 + m) * FEAT + n;
      float x = co[v] + bo;
      float g = cg[v] + bg;
      float silu = x / (1.0f + __expf(-x));
      float sig  = 1.0f / (1.0f + __expf(-g));
      out[idx] = edge_feat[idx] + silu * sig;
    }
  }
}

// ---------------------------------------------------------------------------
extern "C" void kernel_launch(void* const* d_in, const int* in_sizes, int n_in,
                              void* d_out, int out_size, void* d_ws, size_t ws_size,
                              hipStream_t stream) {
  const float* three_basis = (const float*)d_in[0];
  // d_in[1] three_cutoff: dead in reference (weights deleted) -> skipped
  const float* node_feat   = (const float*)d_in[2];
  const float* edge_feat   = (const float*)d_in[3];
  const float* W_atom      = (const float*)d_in[4];
  const float* b_atom      = (const float*)d_in[5];
  const float* W_gate      = (const float*)d_in[6];
  const float* b_gate      = (const float*)d_in[7];
  const float* W_out       = (const float*)d_in[8];
  const float* b_out       = (const float*)d_in[9];
  const int*   edge_dst    = (const int*)d_in[10];
  // d_in[11] lg_src: dead in reference -> skipped
  const int*   lg_dst      = (const int*)d_in[12];
  const int*   segment_ids = (const int*)d_in[13];

  const int n_nodes   = in_sizes[2] / FEAT;
  const int n_edges   = in_sizes[3] / FEAT;
  const int n_triples = in_sizes[13];

  float* atoms = (float*)d_ws;                                   // [n_nodes, 9]
  size_t atoms_elems = (((size_t)n_nodes * NBASIS) + 255) & ~(size_t)255;
  float* nb = atoms + atoms_elems;                               // [n_edges, 9]

  float* out = (float*)d_out;

  // K1: per-atom basis gate
  k_atom_gate<<<(n_nodes + 255) / 256, 256, 0, stream>>>(
      node_feat, W_atom, b_atom, atoms, n_nodes);

  // K2: gather + sorted segmented sum (writes every segment, incl. zeros)
  k_segsum<<<(n_edges + 255) / 256, 256, 0, stream>>>(
      three_basis, atoms, edge_dst, lg_dst, segment_ids, nb, n_edges, n_triples);

  // K3: WMMA gated-MLP bond update + residual
  int tiles = (n_edges + 15) / 16;          // one wave32 per 16-edge tile
  int blocks = (tiles + 7) / 8;             // 8 waves / 256-thread block
  k_bond_update<<<blocks, 256, 0, stream>>>(
      nb, edge_feat, W_out, b_out, W_gate, b_gate, out, n_edges);
}